// NonMaxSuppression_49168785605076
// MI455X (gfx1250) — compile-verified
//
#include <hip/hip_runtime.h>
#include <hip/hip_bf16.h>

// ---------------- constants (match reference) ----------------
#define NB        2
#define NANCH     8400
#define NCH       84
#define NCLS      80
#define CONF_TH   0.25f
#define IOU_TH    0.7f
#define MAX_DET   300
#define CAND      600           // 2*MAX_DET: only first 600 sorted can be kept
#define KPAD      640           // padded to 40 WMMA tiles of 16
#define NTILE     40            // KPAD/16
#define ROWW      20            // u32 words per adjacency row (640 bits)

typedef float v8f __attribute__((ext_vector_type(8)));
typedef float v2f __attribute__((ext_vector_type(2)));

__device__ __forceinline__ unsigned f2ord(float f) {
  unsigned u = __float_as_uint(f);
  return (u & 0x80000000u) ? ~u : (u | 0x80000000u);
}
__device__ __forceinline__ unsigned long long makeKey(unsigned u, int i) {
  return ((unsigned long long)u << 14) | (unsigned)(16383 - i);
}

// ---------------- kernel 1: decode boxes / score / class ----------------
__global__ void nms_decode(const float* __restrict__ pred,
                           float4* __restrict__ wsBox,
                           float*  __restrict__ wsScore,
                           int*    __restrict__ wsCls) {
  int gid = blockIdx.x * blockDim.x + threadIdx.x;
  if (gid >= NB * NANCH) return;
  const float4* p4 = reinterpret_cast<const float4*>(pred + (size_t)gid * NCH);
  float4 g0 = p4[0];                       // cx, cy, w, h
  float hw = g0.z * 0.5f, hh = g0.w * 0.5f;
  float4 box;
  box.x = g0.x - hw; box.y = g0.y - hh;
  box.z = g0.x + hw; box.w = g0.y + hh;
  float best = -INFINITY; int cls = 0;
  #pragma unroll 4
  for (int g = 0; g < NCLS / 4; ++g) {     // classes in order -> first-max argmax
    float4 v = p4[1 + g];
    int k = g * 4;
    if (v.x > best) { best = v.x; cls = k;     }
    if (v.y > best) { best = v.y; cls = k + 1; }
    if (v.z > best) { best = v.z; cls = k + 2; }
    if (v.w > best) { best = v.w; cls = k + 3; }
  }
  wsBox[gid]   = box;
  wsScore[gid] = best;
  wsCls[gid]   = cls;
}

// ---------------- kernel 2: exact top-600 (radix-select + bitonic) ----------------
// one workgroup (1024 threads) per batch
__global__ __launch_bounds__(1024)
void nms_topk(const float4* __restrict__ wsBox,
              const float*  __restrict__ wsScore,
              const int*    __restrict__ wsCls,
              float4* __restrict__ topBox,
              float*  __restrict__ topScore,
              int*    __restrict__ topCls,
              float*  __restrict__ topArea) {
  __shared__ unsigned lu[NANCH];
  __shared__ unsigned hist[256];
  __shared__ unsigned long long cand[1024];
  __shared__ unsigned long long sh_prefix;
  __shared__ int sh_need, sh_cnt;

  const int b   = blockIdx.x;
  const int tid = threadIdx.x;

  // sortable keys: invalid -> -inf (mirrors reference sort key)
  for (int i = tid; i < NANCH; i += 1024) {
    float s  = wsScore[b * NANCH + i];
    float sk = (s > CONF_TH) ? s : -INFINITY;
    lu[i] = f2ord(sk);
  }
  if (tid == 0) { sh_prefix = 0ULL; sh_need = CAND; sh_cnt = 0; }
  __syncthreads();

  // MSB radix-select of the CAND-th largest 46-bit key (keys are all distinct)
  const int shifts[6] = {38, 30, 22, 14, 6, 0};
  const int widths[6] = { 8,  8,  8,  8, 8, 6};
  for (int p = 0; p < 6; ++p) {
    const int shift = shifts[p], width = widths[p];
    if (tid < 256) hist[tid] = 0;
    __syncthreads();
    unsigned long long pref = sh_prefix;
    for (int i = tid; i < NANCH; i += 1024) {
      unsigned long long k = makeKey(lu[i], i);
      if ((k >> (shift + width)) == pref)
        atomicAdd(&hist[(unsigned)(k >> shift) & ((1u << width) - 1u)], 1u);
    }
    __syncthreads();
    if (tid == 0) {
      int need = sh_need, cum = 0, chosen = 0;
      for (int bin = (1 << width) - 1; bin >= 0; --bin) {
        int h = (int)hist[bin];
        if (cum + h >= need) { chosen = bin; sh_need = need - cum; break; }
        cum += h;
      }
      sh_prefix = (pref << width) | (unsigned)chosen;
    }
    __syncthreads();
  }
  const unsigned long long T = sh_prefix;   // exactly CAND keys are >= T

  cand[tid] = 0ULL;
  __syncthreads();
  for (int i = tid; i < NANCH; i += 1024) {
    unsigned long long k = makeKey(lu[i], i);
    if (k >= T) {                            // gather order laundered by sort below
      int p = atomicAdd(&sh_cnt, 1);
      if (p < 1024) cand[p] = k;
    }
  }
  __syncthreads();

  // bitonic sort 1024 keys, descending -> deterministic sorted candidates
  for (unsigned size = 2; size <= 1024; size <<= 1) {
    for (unsigned stride = size >> 1; stride > 0; stride >>= 1) {
      __syncthreads();
      unsigned j = tid ^ stride;
      if (j > (unsigned)tid) {
        unsigned long long a = cand[tid], c = cand[j];
        bool desc = (tid & size) == 0;
        if (desc ? (a < c) : (a > c)) { cand[tid] = c; cand[j] = a; }
      }
    }
  }
  __syncthreads();

  // emit top-CAND (padded to KPAD) gathered candidate data
  for (int t = tid; t < KPAD; t += 1024) {
    float4 box; box.x = 0.f; box.y = 0.f; box.z = 0.f; box.w = 0.f;
    float sc = -INFINITY, ar = 0.f; int cl = 0;
    if (t < CAND) {
      unsigned long long k = cand[t];
      int idx = 16383 - (int)(k & 0x3FFFULL);
      box = wsBox[b * NANCH + idx];
      sc  = wsScore[b * NANCH + idx];
      cl  = wsCls[b * NANCH + idx];
      ar  = (box.z - box.x) * (box.w - box.y);
    }
    int o = b * KPAD + t;
    topBox[o] = box; topScore[o] = sc; topCls[o] = cl; topArea[o] = ar;
  }
}

// ---------------- kernel 3: adjacency bitmask via WMMA + ballot ----------------
// one wave per 16x16 tile; area_i + area_j via V_WMMA_F32_16X16X4_F32
__global__ __launch_bounds__(256)
void nms_adj(const float4* __restrict__ topBox,
             const float*  __restrict__ topArea,
             unsigned short* __restrict__ adj) {
  const int lane = threadIdx.x & 31;
  const int wid  = (blockIdx.x * blockDim.x + threadIdx.x) >> 5;   // 0..3199
  const int b    = wid / (NTILE * NTILE);
  const int t    = wid - b * (NTILE * NTILE);
  const int i0   = (t / NTILE) * 16;
  const int j0   = (t % NTILE) * 16;
  const int m    = lane & 15;
  const int half = lane >> 4;

  const float*  tA = topArea + b * KPAD;
  const float4* tB = topBox  + b * KPAD;

  // A (16x4, M=lane&15): K0=area_i, K1=1 in lanes 0-15; K2=K3=0 in lanes 16-31
  // B (4x16, N=lane&15): K0=1, K1=area_j in lanes 0-15; K2=K3=0 in lanes 16-31
  v2f a, bb;
  float ai = tA[i0 + m];
  float aj = tA[j0 + m];
  a.x  = half ? 0.f : ai;   a.y  = half ? 0.f : 1.f;
  bb.x = half ? 0.f : 1.f;  bb.y = half ? 0.f : aj;
  v8f c = {};
  // c[v] = area_{i0+v+8*half} + area_{j0+(lane&15)}
  c = __builtin_amdgcn_wmma_f32_16x16x4_f32(false, a, false, bb,
                                            (short)0, c, false, false);

  float4 bj = tB[j0 + m];                 // column box, fixed per lane
  #pragma unroll
  for (int v = 0; v < 8; ++v) {
    int i = i0 + v + 8 * half;
    float4 bi = tB[i];
    float iw = fminf(bi.z, bj.z) - fmaxf(bi.x, bj.x);
    float ih = fminf(bi.w, bj.w) - fmaxf(bi.y, bj.y);
    float inter = fmaxf(iw, 0.f) * fmaxf(ih, 0.f);
    float iou   = inter / ((c[v] - inter) + 1e-7f);   // (a_i+a_j - inter) + eps
    unsigned mask = __builtin_amdgcn_ballot_w32(iou > IOU_TH);
    if (lane == 0) {
      int col = j0 >> 4;
      adj[(size_t)(b * KPAD + i0 + v    ) * NTILE + col] = (unsigned short)(mask & 0xFFFFu);
      adj[(size_t)(b * KPAD + i0 + v + 8) * NTILE + col] = (unsigned short)(mask >> 16);
    }
  }
}

// ---------------- kernel 4: serial greedy scan + output ----------------
// one workgroup (256 threads) per batch
__global__ __launch_bounds__(256)
void nms_scan(const unsigned* __restrict__ adj32,
              const float4*  __restrict__ topBox,
              const float*   __restrict__ topScore,
              const int*     __restrict__ topCls,
              float* __restrict__ out) {
  __shared__ unsigned adjL[KPAD * ROWW];   // 51.2 KB
  __shared__ int  vld[KPAD];
  __shared__ int  keepIdx[MAX_DET];
  __shared__ unsigned supp[ROWW];
  __shared__ int  s_cnt;

  const int b   = blockIdx.x;
  const int tid = threadIdx.x;

  for (int w = tid; w < KPAD * ROWW; w += 256)
    adjL[w] = adj32[(size_t)b * KPAD * ROWW + w];
  for (int i = tid; i < KPAD; i += 256)
    vld[i] = topScore[b * KPAD + i] > CONF_TH;
  if (tid < ROWW) supp[tid] = 0u;
  __syncthreads();

  if (tid == 0) {
    int cnt = 0;
    for (int i = 0; i < CAND && cnt < MAX_DET; ++i) {
      if (!vld[i]) continue;
      if ((supp[i >> 5] >> (i & 31)) & 1u) continue;
      keepIdx[cnt++] = i;
      const unsigned* row = &adjL[i * ROWW];
      int wi = i >> 5, bi = i & 31;
      unsigned hiMask = (bi == 31) ? 0u : (0xFFFFFFFFu << (bi + 1)); // only j > i
      supp[wi] |= row[wi] & hiMask;
      for (int w = wi + 1; w < ROWW; ++w) supp[w] |= row[w];
    }
    s_cnt = cnt;
  }
  __syncthreads();

  const int cnt = s_cnt;
  for (int r = tid; r < MAX_DET; r += 256) {
    float o0=0.f,o1=0.f,o2=0.f,o3=0.f,o4=0.f,o5=0.f;
    if (r < cnt) {
      int i = keepIdx[r];
      float4 bx = topBox[b * KPAD + i];
      o0 = bx.x; o1 = bx.y; o2 = bx.z; o3 = bx.w;
      o4 = topScore[b * KPAD + i];
      o5 = (float)topCls[b * KPAD + i];
    }
    float* d = out + (size_t)b * MAX_DET * 6 + (size_t)r * 6;
    d[0]=o0; d[1]=o1; d[2]=o2; d[3]=o3; d[4]=o4; d[5]=o5;
  }
  if (tid == 0)
    reinterpret_cast<int*>(out)[NB * MAX_DET * 6 + b] = cnt;   // int32 num_det bits
}

// ---------------- launcher ----------------
extern "C" void kernel_launch(void* const* d_in, const int* in_sizes, int n_in,
                              void* d_out, int out_size, void* d_ws, size_t ws_size,
                              hipStream_t stream) {
  const float* pred = (const float*)d_in[0];
  char* w = (char*)d_ws;
  // workspace layout (16B-aligned chunks), total 541,440 B
  float4*         wsBox   = (float4*)(w + 0);        // 2*8400 float4  = 268800
  float*          wsScore = (float*) (w + 268800);   // 67200
  int*            wsCls   = (int*)   (w + 336000);   // 67200
  float4*         topBox  = (float4*)(w + 403200);   // 2*640 float4   = 20480
  float*          topScore= (float*) (w + 423680);   // 5120
  int*            topCls  = (int*)   (w + 428800);   // 5120
  float*          topArea = (float*) (w + 433920);   // 5120
  unsigned short* adj     = (unsigned short*)(w + 439040); // 2*640*40*2 = 102400

  // 1) decode: thread per anchor
  {
    int total = NB * NANCH;
    nms_decode<<<(total + 255) / 256, 256, 0, stream>>>(pred, wsBox, wsScore, wsCls);
  }
  // 2) exact top-600 per batch (radix-select + bitonic), one WG per batch
  nms_topk<<<NB, 1024, 0, stream>>>(wsBox, wsScore, wsCls,
                                    topBox, topScore, topCls, topArea);
  // 3) adjacency bitmask: one wave per 16x16 tile; 2*40*40 = 3200 waves
  {
    int waves  = NB * NTILE * NTILE;
    int blocks = waves / 8;                 // 8 waves (256 threads) per block
    nms_adj<<<blocks, 256, 0, stream>>>(topBox, topArea, adj);
  }
  // 4) serial greedy scan + emit detections / num_det
  nms_scan<<<NB, 256, 0, stream>>>((const unsigned*)adj, topBox, topScore, topCls,
                                   (float*)d_out);
}